// RewardNormalizer_25864293056604
// MI455X (gfx1250) — compile-verified
//
#include <hip/hip_runtime.h>
#include <cstdint>

// Problem constants (fixed by the reference).
constexpr int   T_DIM   = 4096;
constexpr int   B_DIM   = 4096;
constexpr int   BQ      = B_DIM / 4;       // float4 columns = 1024
constexpr int   CHUNKS  = 64;              // chunks along T
constexpr int   CLEN    = T_DIM / CHUNKS;  // 64 rows per chunk
constexpr int   CB      = CHUNKS * B_DIM;  // 262144 coeffs per array
constexpr float GAMMA_F = 0.99f;
constexpr float EPS_F   = 1e-8f;
constexpr float CLIP_F  = 10.0f;
constexpr float VMOM_F  = 0.99f;

typedef __attribute__((ext_vector_type(2))) float v2f;
typedef __attribute__((ext_vector_type(8))) float v8f;

// ---------------------------------------------------------------------------
// Pass 1: per-(column,chunk) affine scan coefficients + moment coefficients.
// Reverse scan within chunk: after processing row t, value_t = A_t*x + B_t
// where x is the carry entering the chunk from higher t.
// Stores 7 arrays [c*B + j] into scratch (aliased onto d_out).
// ---------------------------------------------------------------------------
__global__ __launch_bounds__(256) void k1_chunk_coeffs(
    const float* __restrict__ rewards,
    const float* __restrict__ dones,
    float* __restrict__ w) {
  const int gid = blockIdx.x * 256 + threadIdx.x;   // 0 .. 65535
  const int jq  = gid & (BQ - 1);                   // float4 column
  const int c   = gid >> 10;                        // chunk id

  const float4* r4 = reinterpret_cast<const float4*>(rewards);
  const float4* d4 = reinterpret_cast<const float4*>(dones);

  float A[4]  = {1.f, 1.f, 1.f, 1.f};
  float Bv[4] = {0.f, 0.f, 0.f, 0.f};
  float SA[4] = {0.f, 0.f, 0.f, 0.f};
  float SB[4] = {0.f, 0.f, 0.f, 0.f};
  float SA2[4] = {0.f, 0.f, 0.f, 0.f};
  float SAB[4] = {0.f, 0.f, 0.f, 0.f};
  float SB2[4] = {0.f, 0.f, 0.f, 0.f};

  const int top = (c + 1) * CLEN;
  const int bot = c * CLEN;
  for (int t = top - 1; t >= bot; --t) {
    float4 r = r4[t * BQ + jq];
    float4 d = d4[t * BQ + jq];
    float rr[4] = {r.x, r.y, r.z, r.w};
    float dd[4] = {d.x, d.y, d.z, d.w};
#pragma unroll
    for (int e = 0; e < 4; ++e) {
      float a = GAMMA_F * (1.0f - dd[e]);
      A[e]  = a * A[e];
      Bv[e] = fmaf(a, Bv[e], rr[e]);
      SA[e]  += A[e];
      SB[e]  += Bv[e];
      SA2[e] = fmaf(A[e],  A[e],  SA2[e]);
      SAB[e] = fmaf(A[e],  Bv[e], SAB[e]);
      SB2[e] = fmaf(Bv[e], Bv[e], SB2[e]);
    }
  }

  const int o4 = c * BQ + jq;
  float4* wA   = reinterpret_cast<float4*>(w + 0 * CB);
  float4* wB   = reinterpret_cast<float4*>(w + 1 * CB);
  float4* wSA  = reinterpret_cast<float4*>(w + 2 * CB);
  float4* wSB  = reinterpret_cast<float4*>(w + 3 * CB);
  float4* wSA2 = reinterpret_cast<float4*>(w + 4 * CB);
  float4* wSAB = reinterpret_cast<float4*>(w + 5 * CB);
  float4* wSB2 = reinterpret_cast<float4*>(w + 6 * CB);
  wA[o4]   = make_float4(A[0], A[1], A[2], A[3]);
  wB[o4]   = make_float4(Bv[0], Bv[1], Bv[2], Bv[3]);
  wSA[o4]  = make_float4(SA[0], SA[1], SA[2], SA[3]);
  wSB[o4]  = make_float4(SB[0], SB[1], SB[2], SB[3]);
  wSA2[o4] = make_float4(SA2[0], SA2[1], SA2[2], SA2[3]);
  wSAB[o4] = make_float4(SAB[0], SAB[1], SAB[2], SAB[3]);
  wSB2[o4] = make_float4(SB2[0], SB2[1], SB2[2], SB2[3]);
}

// ---------------------------------------------------------------------------
// Wave-level deterministic reduction of two values via V_WMMA_F32_16X16X4_F32.
// A: a0 = s (K=0 lanes 0-15 / K=2 lanes 16-31), a1 = q (K=1 / K=3).
// B: column 0 selects s terms (B[0][0]=B[2][0]=1), column 1 selects q terms.
// D[m][0] = s_m + s_{m+16};  D[m][1] = q_m + q_{m+16}.
// D VGPR r: lanes 0-15 -> M=r, lanes 16-31 -> M=8+r, N = lane%16.
// ---------------------------------------------------------------------------
__device__ __forceinline__ void wave_reduce2_wmma(float s, float q,
                                                  float* ts, float* tq) {
  const unsigned lane = threadIdx.x & 31u;
  v2f a; a.x = s; a.y = q;
  v2f b;
  b.x = (lane == 0u || lane == 16u) ? 1.0f : 0.0f;
  b.y = (lane == 1u || lane == 17u) ? 1.0f : 0.0f;
  v8f c = {};
  c = __builtin_amdgcn_wmma_f32_16x16x4_f32(false, a, false, b,
                                            (short)0, c, false, false);
  float part = c[0] + c[1] + c[2] + c[3] + c[4] + c[5] + c[6] + c[7];
  // lane0: sum_{m=0..7} (s_m+s_{m+16}); lane16: rows 8..15. lanes 1/17: q.
  *ts = __shfl(part, 0, 32) + __shfl(part, 16, 32);
  *tq = __shfl(part, 1, 32) + __shfl(part, 17, 32);
}

// ---------------------------------------------------------------------------
// Pass 2: per-column serial combine over 64 chunks (reverse order), producing
// per-block (sum, sumsq) partials, reduced deterministically.
// ---------------------------------------------------------------------------
__global__ __launch_bounds__(256) void k2_combine(
    const float* __restrict__ w,
    const float* __restrict__ return_init,
    float* __restrict__ ws) {
  const int j = blockIdx.x * 256 + threadIdx.x;    // column 0..4095
  const float* cA   = w + 0 * CB;
  const float* cBv  = w + 1 * CB;
  const float* cSA  = w + 2 * CB;
  const float* cSB  = w + 3 * CB;
  const float* cSA2 = w + 4 * CB;
  const float* cSAB = w + 5 * CB;
  const float* cSB2 = w + 6 * CB;

  float x = return_init[0];
  float s = 0.0f, q = 0.0f;
  for (int c = CHUNKS - 1; c >= 0; --c) {
    const int idx = c * B_DIM + j;
    float sa  = cSA[idx],  sb  = cSB[idx];
    float sa2 = cSA2[idx], sab = cSAB[idx], sb2 = cSB2[idx];
    float Ac  = cA[idx],   Bc  = cBv[idx];
    s += fmaf(sa, x, sb);
    q += fmaf(sa2, x * x, fmaf(2.0f * sab, x, sb2));
    x = fmaf(Ac, x, Bc);
  }

  float ts, tq;
  wave_reduce2_wmma(s, q, &ts, &tq);

  __shared__ float lds[16];
  const int wv = threadIdx.x >> 5;
  if ((threadIdx.x & 31) == 0) { lds[2 * wv] = ts; lds[2 * wv + 1] = tq; }
  __syncthreads();
  if (threadIdx.x == 0) {
    float bs = 0.0f, bq = 0.0f;
#pragma unroll
    for (int i = 0; i < 8; ++i) { bs += lds[2 * i]; bq += lds[2 * i + 1]; }
    ws[8 + 2 * blockIdx.x]     = bs;   // deterministic per-block partials
    ws[8 + 2 * blockIdx.x + 1] = bq;
  }
}

// ---------------------------------------------------------------------------
// Pass 3: fold 16 block partials (fixed order), compute unbiased variance,
// EMA-update, and the final normalization scale.
// ---------------------------------------------------------------------------
__global__ void k3_finalize(const float* __restrict__ var_init,
                            float* __restrict__ ws) {
  float s = 0.0f, q = 0.0f;
#pragma unroll
  for (int i = 0; i < 16; ++i) { s += ws[8 + 2 * i]; q += ws[8 + 2 * i + 1]; }
  const float n    = 16777216.0f;
  const float mean = s / n;
  const float var  = (q - s * mean) / (n - 1.0f);
  const float vnew = var_init[0] * VMOM_F + var * (1.0f - VMOM_F);
  ws[0] = 1.0f / sqrtf(vnew + EPS_F);
}

// ---------------------------------------------------------------------------
// Pass 4: normalize + clip. Streams rewards via the CDNA5 async-to-LDS path
// (global_load_async_to_lds_b128, ASYNCcnt). Each lane loads its own 16B LDS
// slot, so a wave-local s_wait_asynccnt suffices before the ds read.
// ---------------------------------------------------------------------------
__global__ __launch_bounds__(256) void k4_normalize(
    const float* __restrict__ rewards,
    const float* __restrict__ ws,
    float* __restrict__ out) {
  __shared__ float4 tile[256];
  const unsigned gid = blockIdx.x * 256u + threadIdx.x;
  const float4* r4 = reinterpret_cast<const float4*>(rewards);

  const float4* gsrc = r4 + gid;
  unsigned lds_off =
      (unsigned)(unsigned long long)(uintptr_t)(&tile[threadIdx.x]);
  asm volatile("global_load_async_to_lds_b128 %0, %1, off"
               :
               : "v"(lds_off), "v"(gsrc)
               : "memory");
  asm volatile("s_wait_asynccnt 0" ::: "memory");

  const float scale = ws[0];
  float4 v = tile[threadIdx.x];
  float4 o;
  o.x = fminf(fmaxf(v.x * scale, -CLIP_F), CLIP_F);
  o.y = fminf(fmaxf(v.y * scale, -CLIP_F), CLIP_F);
  o.z = fminf(fmaxf(v.z * scale, -CLIP_F), CLIP_F);
  o.w = fminf(fmaxf(v.w * scale, -CLIP_F), CLIP_F);
  reinterpret_cast<float4*>(out)[gid] = o;
}

// ---------------------------------------------------------------------------
extern "C" void kernel_launch(void* const* d_in, const int* in_sizes, int n_in,
                              void* d_out, int out_size, void* d_ws,
                              size_t ws_size, hipStream_t stream) {
  (void)in_sizes; (void)n_in; (void)out_size; (void)ws_size;
  const float* rewards  = (const float*)d_in[0];
  const float* dones    = (const float*)d_in[1];
  const float* ret_init = (const float*)d_in[2];
  const float* var_init = (const float*)d_in[3];
  float* out = (float*)d_out;
  float* ws  = (float*)d_ws;

  // Coefficient scratch (7 * 1 MB) lives in d_out; fully overwritten by k4.
  float* scratch = out;

  k1_chunk_coeffs<<<(CHUNKS * BQ) / 256, 256, 0, stream>>>(rewards, dones,
                                                           scratch);
  k2_combine<<<B_DIM / 256, 256, 0, stream>>>(scratch, ret_init, ws);
  k3_finalize<<<1, 1, 0, stream>>>(var_init, ws);
  k4_normalize<<<(T_DIM * B_DIM / 4) / 256, 256, 0, stream>>>(rewards, ws,
                                                              out);
}